// RelativePositionAttention_54889682043142
// MI455X (gfx1250) — compile-verified
//
#include <hip/hip_runtime.h>
#include <math.h>

typedef float v2f __attribute__((ext_vector_type(2)));
typedef float v4fv __attribute__((ext_vector_type(4)));
typedef float v8f __attribute__((ext_vector_type(8)));

constexpr int DM  = 1024;  // D_MODEL
constexpr int NH  = 16;    // heads
constexpr int DH  = 64;    // head dim
constexpr int SEQ = 1024;  // T
constexpr int NB  = 4;     // batch
constexpr int MAXLEN = 1024;

#ifndef __has_builtin
#define __has_builtin(x) 0
#endif

#if __has_builtin(__builtin_amdgcn_global_load_async_to_lds_b128)
#define HAVE_ASYNC_LDS 1
#else
#define HAVE_ASYNC_LDS 0
#endif

// Exact parameter types per hipcc diagnostic:
//   param 1: int __vector(4) addrspace(1)*  (global source)
typedef int v4i_b128 __attribute__((vector_size(16)));
typedef __attribute__((address_space(1))) v4i_b128* async_gptr;
typedef __attribute__((address_space(3))) v4i_b128* async_lptr;

// Async global->LDS 16B copy (ASYNCcnt-tracked) with synchronous fallback.
__device__ __forceinline__ void g2l_b128(float* lds_dst, const float* gsrc) {
#if HAVE_ASYNC_LDS
  __builtin_amdgcn_global_load_async_to_lds_b128(
      (async_gptr)gsrc, (async_lptr)lds_dst, 0, 0);
#else
  *(v4fv*)lds_dst = *(const v4fv*)gsrc;
#endif
}

template <int N>
__device__ __forceinline__ void wait_async() {
#if HAVE_ASYNC_LDS
#if __has_builtin(__builtin_amdgcn_s_wait_asynccnt)
  __builtin_amdgcn_s_wait_asynccnt(N);
#else
  asm volatile("s_wait_asynccnt %0" ::"i"(N) : "memory");
#endif
#endif
}

// D = A(16x4,f32) * B(4x16,f32) + C(16x16,f32), fp32 matrix pipe.
__device__ __forceinline__ v8f wmma_f32(v2f a, v2f b, v8f c) {
  return __builtin_amdgcn_wmma_f32_16x16x4_f32(false, a, false, b, (short)0, c,
                                               false, false);
}

// ---------------------------------------------------------------------------
// Kernel 1: out[b,h,t,dh] = x @ W + bias (head-major output).
// Block: 256 thr = 8 waves; wave tile 32x64; block tile 256x64.
// W K-panel (32x64) staged in LDS (double-buffered, async), stride 72 dwords
// so each B-fragment ds read hits 32 distinct banks.
// ---------------------------------------------------------------------------
constexpr int KC = 32;     // K-panel depth
constexpr int BSTR = 72;   // LDS row stride (floats) for W panel

__global__ __launch_bounds__(256) void qkv_proj_kernel(
    const float* __restrict__ x, const float* __restrict__ W,
    const float* __restrict__ bias, float* __restrict__ out) {
  __shared__ float sB[2][KC * BSTR];

  const int tid = threadIdx.x;
  const int lane = tid & 31;
  const int wave = tid >> 5;
  const int lo = lane & 15;
  const int hi = lane >> 4;
  const int row0 = blockIdx.y * 256 + wave * 32;  // global row = b*SEQ + t
  const int col0 = blockIdx.x * 64;

  auto stage = [&](int buf, int k0) {
    // 32 rows x 64 cols = 512 16B-chunks; 256 threads x 2.
#pragma unroll
    for (int rr = 0; rr < 2; ++rr) {
      const int idx = tid + rr * 256;
      const int row = idx >> 4;
      const int c4 = (idx & 15) * 4;
      g2l_b128(&sB[buf][row * BSTR + c4],
               W + (size_t)(k0 + row) * DM + col0 + c4);
    }
  };

  v8f acc[8] = {};  // [rt*4 + t]
  const float* xrow0 = x + (size_t)(row0 + lo) * DM;
  const float* xrow1 = x + (size_t)(row0 + 16 + lo) * DM;

  constexpr int NC = DM / KC;  // 32 chunks
  stage(0, 0);
  for (int c = 0; c < NC; ++c) {
    if (c + 1 < NC) {
      stage((c + 1) & 1, (c + 1) * KC);
      wait_async<2>();  // keep only the just-issued panel in flight
    } else {
      wait_async<0>();
    }
    __syncthreads();
    const float* bb = sB[c & 1];
    const int kbase = c * KC;
#pragma unroll
    for (int kl = 0; kl < KC; kl += 4) {
      v2f a0 = *(const v2f*)(xrow0 + kbase + kl + 2 * hi);
      v2f a1 = *(const v2f*)(xrow1 + kbase + kl + 2 * hi);
#pragma unroll
      for (int t = 0; t < 4; ++t) {
        v2f bf;
        bf.x = bb[(kl + 2 * hi) * BSTR + t * 16 + lo];
        bf.y = bb[(kl + 2 * hi + 1) * BSTR + t * 16 + lo];
        acc[t] = wmma_f32(a0, bf, acc[t]);
        acc[4 + t] = wmma_f32(a1, bf, acc[4 + t]);
      }
    }
    __syncthreads();
  }

#pragma unroll
  for (int rt = 0; rt < 2; ++rt) {
#pragma unroll
    for (int t = 0; t < 4; ++t) {
      const int col = col0 + t * 16 + lo;
      const float bv = bias[col];
      const int h = col >> 6, dh = col & (DH - 1);
#pragma unroll
      for (int r = 0; r < 8; ++r) {
        const int row = row0 + rt * 16 + r + 8 * hi;
        const int b = row >> 10, tt = row & (SEQ - 1);
        out[((size_t)(b * NH + h) * SEQ + tt) * DH + dh] =
            acc[rt * 4 + t][r] + bv;
      }
    }
  }
}

// ---------------------------------------------------------------------------
// Kernel 2: QEr[bh,i,s] = q[bh,i,:] . Er_t[s,:]   (K = 64)
// Er is 256KB and L2-hot; direct b64 fragment loads are fine.
// ---------------------------------------------------------------------------
__global__ __launch_bounds__(256) void qer_kernel(
    const float* __restrict__ q, const float* __restrict__ er,
    float* __restrict__ qer) {
  const int lane = threadIdx.x & 31;
  const int wave = threadIdx.x >> 5;
  const int lo = lane & 15, hi = lane >> 4;
  const int bh = blockIdx.z;
  const int i0 = blockIdx.y * 128 + wave * 16;
  const int s0 = blockIdx.x * 64;

  const float* qrow = q + ((size_t)bh * SEQ + i0 + lo) * DH;
  v8f acc[4] = {};
#pragma unroll
  for (int k = 0; k < DH; k += 4) {
    v2f a = *(const v2f*)(qrow + k + 2 * hi);
#pragma unroll
    for (int t = 0; t < 4; ++t) {
      v2f bf = *(const v2f*)(er + (size_t)(s0 + t * 16 + lo) * DH + k + 2 * hi);
      acc[t] = wmma_f32(a, bf, acc[t]);
    }
  }
  float* orow = qer + (size_t)bh * SEQ * SEQ;
#pragma unroll
  for (int t = 0; t < 4; ++t)
#pragma unroll
    for (int r = 0; r < 8; ++r)
      orow[(size_t)(i0 + r + 8 * hi) * SEQ + s0 + t * 16 + lo] = acc[t][r];
}

// ---------------------------------------------------------------------------
// Kernel 3: flash attention with relative bias.
// Block: 128 thr = 4 waves, 16 rows/wave -> 64 rows/block, one (b,h).
// K/V 16x64 tiles staged in LDS (double-buffered, async), shared by 4 waves.
// Strides: K=68 (conflict-free b64 frag reads), V=72 (conflict-free b32).
// ---------------------------------------------------------------------------
constexpr int KSTR = 68;
constexpr int VSTR = 72;

__global__ __launch_bounds__(128) void attn_kernel(
    const float* __restrict__ q, const float* __restrict__ kmat,
    const float* __restrict__ vmat, const float* __restrict__ qer,
    float* __restrict__ out) {
  __shared__ float sK[2][16 * KSTR];
  __shared__ float sV[2][16 * VSTR];
  __shared__ float ldsP[4][16 * 17];

  const int tid = threadIdx.x;
  const int lane = tid & 31;
  const int wave = tid >> 5;
  const int lo = lane & 15, hi = lane >> 4;
  const int h = blockIdx.y, b = blockIdx.z;
  const int bh = b * NH + h;
  const int iblock = blockIdx.x * 64;
  const int i0 = iblock + wave * 16;

  const float* qb = q + (size_t)bh * SEQ * DH;
  const float* kb = kmat + (size_t)bh * SEQ * DH;
  const float* vb = vmat + (size_t)bh * SEQ * DH;
  const float* qerb = qer + (size_t)bh * SEQ * SEQ;

  auto stageKV = [&](int buf, int j0s) {
    // 2 x (16 rows x 64 cols) = 2 x 256 16B-chunks; 128 threads x 2 each.
#pragma unroll
    for (int rr = 0; rr < 2; ++rr) {
      const int idx = tid + rr * 128;
      const int row = idx >> 4;
      const int c4 = (idx & 15) * 4;
      g2l_b128(&sK[buf][row * KSTR + c4], kb + (size_t)(j0s + row) * DH + c4);
      g2l_b128(&sV[buf][row * VSTR + c4], vb + (size_t)(j0s + row) * DH + c4);
    }
  };

  // Preload Q A-fragments for this wave's 16 rows (K = 64 -> 16 steps).
  v2f qf[16];
  {
    const float* qrow = qb + (size_t)(i0 + lo) * DH;
#pragma unroll
    for (int kk = 0; kk < 16; ++kk)
      qf[kk] = *(const v2f*)(qrow + kk * 4 + 2 * hi);
  }

  v8f o[4] = {};
  float mrow[8], lrow[8];
#pragma unroll
  for (int r = 0; r < 8; ++r) { mrow[r] = -INFINITY; lrow[r] = 0.0f; }

  const int jbEnd = (iblock + 63) >> 4;  // uniform across the block
  stageKV(0, 0);
  for (int jb = 0; jb <= jbEnd; ++jb) {
    const int j0 = jb * 16;
    if (jb < jbEnd) {
      stageKV((jb + 1) & 1, j0 + 16);
      wait_async<4>();  // retire current tile, keep next in flight
    } else {
      wait_async<0>();
    }
    __syncthreads();
    const float* kt = sK[jb & 1];
    const float* vt = sV[jb & 1];

    // S = Q * K^T : B[K=d, N=j] from LDS.
    v8f s = {};
#pragma unroll
    for (int kk = 0; kk < 16; ++kk) {
      v2f bf = *(const v2f*)(kt + lo * KSTR + kk * 4 + 2 * hi);
      s = wmma_f32(qf[kk], bf, s);
    }

    // Relative bias (skewed gather), scale, causal mask, online softmax.
    float p[8];
#pragma unroll
    for (int r = 0; r < 8; ++r) {
      const int i = i0 + r + 8 * hi;
      const int j = j0 + lo;
      float val = -INFINITY;
      if (j <= i) {
        const float rel = qerb[(size_t)i * SEQ + (SEQ - 1 - i + j)];
        val = (s[r] + rel) * 0.125f;  // 1/sqrt(DH)
      }
      float rm = val;
#pragma unroll
      for (int off = 8; off >= 1; off >>= 1)
        rm = fmaxf(rm, __shfl_xor(rm, off, 16));
      const float mnew = fmaxf(mrow[r], rm);
      const float alpha = __expf(mrow[r] - mnew);
      const float pe = __expf(val - mnew);
      float rs = pe;
#pragma unroll
      for (int off = 8; off >= 1; off >>= 1) rs += __shfl_xor(rs, off, 16);
      lrow[r] = lrow[r] * alpha + rs;
      mrow[r] = mnew;
#pragma unroll
      for (int t = 0; t < 4; ++t) o[t][r] *= alpha;
      p[r] = pe;
    }

    // C-layout P -> LDS -> A-layout fragments.
#pragma unroll
    for (int r = 0; r < 8; ++r)
      ldsP[wave][(r + 8 * hi) * 17 + lo] = p[r];
    __syncthreads();

    // O += P * V (K = 16 -> 4 steps), V fragments from LDS.
#pragma unroll
    for (int kk = 0; kk < 4; ++kk) {
      v2f af;
      af.x = ldsP[wave][lo * 17 + kk * 4 + 2 * hi];
      af.y = ldsP[wave][lo * 17 + kk * 4 + 2 * hi + 1];
#pragma unroll
      for (int t = 0; t < 4; ++t) {
        v2f bf;
        bf.x = vt[(kk * 4 + 2 * hi) * VSTR + t * 16 + lo];
        bf.y = vt[(kk * 4 + 2 * hi + 1) * VSTR + t * 16 + lo];
        o[t] = wmma_f32(af, bf, o[t]);
      }
    }
    __syncthreads();  // protect sK/sV buffer reuse and ldsP for next iter
  }

  // Finalize: divide by row sum, write out[b, t, h*DH + d].
#pragma unroll
  for (int r = 0; r < 8; ++r) {
    const float inv = 1.0f / lrow[r];
    const int i = i0 + r + 8 * hi;
#pragma unroll
    for (int t = 0; t < 4; ++t)
      out[((size_t)b * SEQ + i) * DM + h * DH + t * 16 + lo] = o[t][r] * inv;
  }
}

// ---------------------------------------------------------------------------
extern "C" void kernel_launch(void* const* d_in, const int* in_sizes, int n_in,
                              void* d_out, int out_size, void* d_ws,
                              size_t ws_size, hipStream_t stream) {
  (void)in_sizes; (void)n_in; (void)out_size; (void)ws_size;
  const float* x  = (const float*)d_in[0];
  const float* Wq = (const float*)d_in[1];
  const float* bq = (const float*)d_in[2];
  const float* Wk = (const float*)d_in[3];
  const float* bk = (const float*)d_in[4];
  const float* Wv = (const float*)d_in[5];
  const float* bv = (const float*)d_in[6];
  const float* Er = (const float*)d_in[7];
  float* out = (float*)d_out;
  float* ws = (float*)d_ws;

  const size_t QKV_ELEMS = (size_t)NB * NH * SEQ * DH;  // 4.19M floats each
  float* qd   = ws;
  float* kd   = ws + QKV_ELEMS;
  float* vd   = ws + 2 * QKV_ELEMS;
  float* qerd = ws + 3 * QKV_ELEMS;  // NB*NH*SEQ*SEQ floats (268 MB)

  dim3 gProj(DM / 64, (NB * SEQ) / 256, 1);
  qkv_proj_kernel<<<gProj, 256, 0, stream>>>(x, Wq, bq, qd);
  qkv_proj_kernel<<<gProj, 256, 0, stream>>>(x, Wk, bk, kd);
  qkv_proj_kernel<<<gProj, 256, 0, stream>>>(x, Wv, bv, vd);

  const float* ErT = Er + (size_t)(MAXLEN - SEQ) * DH;  // == Er for T==1024
  dim3 gQer(SEQ / 64, SEQ / 128, NB * NH);
  qer_kernel<<<gQer, 256, 0, stream>>>(qd, ErT, qerd);

  dim3 gAttn(SEQ / 64, NH, NB);
  attn_kernel<<<gAttn, 128, 0, stream>>>(qd, kd, vd, qerd, out);
}